// HierarchicalECPredictor_17188459119287
// MI455X (gfx1250) — compile-verified
//
#include <hip/hip_runtime.h>
#include <hip/hip_bf16.h>
#include <math.h>

typedef __attribute__((ext_vector_type(2))) float v2f;
typedef __attribute__((ext_vector_type(4))) float v4f;
typedef __attribute__((ext_vector_type(8))) float v8f;

#define GRAPHS 1024
#define HID 256

__device__ __forceinline__ float gelu_exact(float x) {
    return 0.5f * x * (1.0f + erff(x * 0.70710678118654752f));
}

__device__ __forceinline__ int lower_bound_i32(const int* a, int n, int v) {
    int lo = 0, hi = n;
    while (lo < hi) { int mid = (lo + hi) >> 1; if (a[mid] < v) lo = mid + 1; else hi = mid; }
    return lo;
}

// --- Stage 1: segment mean (batch_idx sorted). One block per graph.
// 256 threads = 64 float4 column-groups x 4 row-slices; each thread streams
// nontemporal 128-bit loads (global_load_b128, TH=NT: data is touched once,
// don't churn the 192MB L2). LDS-reduce the 4 row-slices, then mean.
// This kernel is the roofline: ~410 MB read at 23.3 TB/s.
__global__ void seg_mean_kernel(const float* __restrict__ x, const int* __restrict__ batch,
                                int n_nodes, float* __restrict__ gmean) {
    __shared__ v4f part[256];
    int g  = blockIdx.x;
    int t  = threadIdx.x;
    int c4 = t & 63;        // float4 column group: cols 4*c4 .. 4*c4+3
    int rs = t >> 6;        // row slice 0..3
    int lo = lower_bound_i32(batch, n_nodes, g);
    int hi = lower_bound_i32(batch, n_nodes, g + 1);

    const v4f* x4 = (const v4f*)x;  // HID/4 = 64 vectors per row
    v4f acc = {0.f, 0.f, 0.f, 0.f};
    for (int r = lo + rs; r < hi; r += 4) {
        v4f v = __builtin_nontemporal_load(&x4[(size_t)r * 64 + c4]);
        acc += v;
    }
    part[t] = acc;
    __syncthreads();
    if (t < 64) {
        v4f s = part[t] + part[t + 64] + part[t + 128] + part[t + 192];
        float inv = 1.0f / fmaxf((float)(hi - lo), 1.0f);
        s *= inv;
        ((v4f*)gmean)[(size_t)g * 64 + t] = s;
    }
}

// --- Stage 2: fp32 WMMA GEMM: Y = act(X[MxK] @ W[KxN] + bias), row-major.
// Requires M%16==0, N%16==0, K%4==0. One wave per 16x16 output tile.
// A layout (ISA 32-bit A 16x4): lanes 0-15 -> M=lane, V0=K+0, V1=K+1;
//                               lanes 16-31 -> M=lane-16, V0=K+2, V1=K+3.
// C/D layout: lanes 0-15 -> M=r, lanes 16-31 -> M=8+r, N=lane%16.
// Optionally mirrors the activated result into Y2/Y3 (concat buffers).
__global__ void wmma_gemm_f32(const float* __restrict__ X, int ldx,
                              const float* __restrict__ W,
                              const float* __restrict__ bias,
                              float* __restrict__ Y, int ldy,
                              float* __restrict__ Y2, int ldy2,
                              float* __restrict__ Y3, int ldy3,
                              int M, int N, int K, int act) {
    const int wavesPerBlock = blockDim.x >> 5;
    int wave = blockIdx.x * wavesPerBlock + (threadIdx.x >> 5);
    int tilesN = N >> 4;
    int tileM = wave / tilesN, tileN = wave % tilesN;
    if ((tileM << 4) >= M) return;                // wave-uniform branch
    int lane = threadIdx.x & 31;
    int half = lane >> 4, lr = lane & 15;
    int m0 = tileM << 4, n0 = tileN << 4;

    float bv = bias[n0 + lr];
    v8f acc;
#pragma unroll
    for (int i = 0; i < 8; ++i) acc[i] = bv;      // bias broadcast along M

    const float* xrow = X + (size_t)(m0 + lr) * ldx + 2 * half;
    const float* wcol = W + (size_t)(2 * half) * N + n0 + lr;
    for (int k = 0; k < K; k += 4) {
        v2f a, b;
        a[0] = xrow[k + 0];
        a[1] = xrow[k + 1];
        const float* wp = wcol + (size_t)k * N;
        b[0] = wp[0];
        b[1] = wp[N];
        acc = __builtin_amdgcn_wmma_f32_16x16x4_f32(
            false, a, false, b, (short)0, acc, false, false);
    }
#pragma unroll
    for (int r2 = 0; r2 < 8; ++r2) {
        int m = m0 + r2 + 8 * half;
        float v = acc[r2];
        if (act) v = gelu_exact(v);
        Y[(size_t)m * ldy + n0 + lr] = v;
        if (Y2) Y2[(size_t)m * ldy2 + n0 + lr] = v;
        if (Y3) Y3[(size_t)m * ldy3 + n0 + lr] = v;
    }
}

// --- Stage 3: classifier head: logits = h[Gx128] @ w2[128xE] + b2, optional
// row softmax. One block (128 threads) per graph; h row cached in LDS.
__global__ void head_kernel(const float* __restrict__ h,
                            const float* __restrict__ w2,
                            const float* __restrict__ b2, int E,
                            float* __restrict__ logits, int ldl,
                            int do_softmax,
                            float* __restrict__ probs_out, int ldp,
                            float* __restrict__ cat_a, int lda, int offa,
                            float* __restrict__ cat_b, int ldb, int offb) {
    __shared__ float hrow[128];
    __shared__ float lbuf[304];
    __shared__ float red[128];
    int g = blockIdx.x;
    int t = threadIdx.x;
    hrow[t] = h[(size_t)g * 128 + t];
    __syncthreads();
    for (int e = t; e < E; e += 128) {
        float acc = b2[e];
#pragma unroll 4
        for (int k = 0; k < 128; ++k) acc += hrow[k] * w2[(size_t)k * E + e];
        lbuf[e] = acc;
        logits[(size_t)g * ldl + e] = acc;
    }
    if (!do_softmax) return;
    __syncthreads();
    float m = -INFINITY;
    for (int e = t; e < E; e += 128) m = fmaxf(m, lbuf[e]);
    red[t] = m;
    __syncthreads();
    for (int s = 64; s > 0; s >>= 1) { if (t < s) red[t] = fmaxf(red[t], red[t + s]); __syncthreads(); }
    float mx = red[0];
    __syncthreads();
    float sum = 0.f;
    for (int e = t; e < E; e += 128) { float ev = expf(lbuf[e] - mx); lbuf[e] = ev; sum += ev; }
    red[t] = sum;
    __syncthreads();
    for (int s = 64; s > 0; s >>= 1) { if (t < s) red[t] += red[t + s]; __syncthreads(); }
    float inv = 1.0f / red[0];
    for (int e = t; e < E; e += 128) {
        float p = lbuf[e] * inv;
        if (probs_out) probs_out[(size_t)g * ldp + e] = p;
        if (cat_a) cat_a[(size_t)g * lda + offa + e] = p;
        if (cat_b) cat_b[(size_t)g * ldb + offb + e] = p;
    }
}

// --- Stage 0: build zero-padded weight copies (K padded to %4) and zero the
// pad columns of the concat buffers x2 (264) / x3 (336).
__global__ void prep_kernel(const float* __restrict__ w21, const float* __restrict__ w31,
                            float* __restrict__ wp2, float* __restrict__ wp3,
                            float* __restrict__ x2, float* __restrict__ x3) {
    int i = blockIdx.x * blockDim.x + threadIdx.x;
    const int N2 = 264 * 128, N3 = 336 * 128;
    if (i < N2) {
        int r = i / 128, c = i % 128;
        wp2[i] = (r < 263) ? w21[r * 128 + c] : 0.f;
    } else if (i < N2 + N3) {
        int j = i - N2;
        int r = j / 128, c = j % 128;
        wp3[j] = (r < 333) ? w31[r * 128 + c] : 0.f;
    } else if (i < N2 + N3 + GRAPHS) {
        int g = i - (N2 + N3);
        x2[(size_t)g * 264 + 263] = 0.f;
    } else if (i < N2 + N3 + GRAPHS + 3 * GRAPHS) {
        int j = i - (N2 + N3 + GRAPHS);
        int g = j / 3, c = j % 3;
        x3[(size_t)g * 336 + 333 + c] = 0.f;
    }
}

extern "C" void kernel_launch(void* const* d_in, const int* in_sizes, int n_in,
                              void* d_out, int out_size, void* d_ws, size_t ws_size,
                              hipStream_t stream) {
    const float* node   = (const float*)d_in[0];
    const int*   batch  = (const int*)  d_in[1];
    const float* pool_w = (const float*)d_in[2];
    const float* pool_b = (const float*)d_in[3];
    const float* ec1_w1 = (const float*)d_in[4];
    const float* ec1_b1 = (const float*)d_in[5];
    const float* ec1_w2 = (const float*)d_in[6];
    const float* ec1_b2 = (const float*)d_in[7];
    const float* ec2_w1 = (const float*)d_in[8];
    const float* ec2_b1 = (const float*)d_in[9];
    const float* ec2_w2 = (const float*)d_in[10];
    const float* ec2_b2 = (const float*)d_in[11];
    const float* ec3_w1 = (const float*)d_in[12];
    const float* ec3_b1 = (const float*)d_in[13];
    const float* ec3_w2 = (const float*)d_in[14];
    const float* ec3_b2 = (const float*)d_in[15];
    int n_nodes = in_sizes[0] / HID;

    float* out     = (float*)d_out;
    float* out_ec1 = out;                          // [1024 x 7]
    float* out_ec2 = out + GRAPHS * 7;             // [1024 x 70]
    float* out_ec3 = out + GRAPHS * (7 + 70);      // [1024 x 300]
    float* out_p1  = out + GRAPHS * (7 + 70 + 300);      // [1024 x 7]
    float* out_g   = out + GRAPHS * (7 + 70 + 300 + 7);  // [1024 x 256]

    float* ws    = (float*)d_ws;
    float* gmean = ws;                             // 1024*256
    float* hbuf  = gmean + GRAPHS * HID;           // 1024*128 (reused 3x)
    float* x2    = hbuf + GRAPHS * 128;            // 1024*264
    float* x3    = x2 + GRAPHS * 264;              // 1024*336
    float* wp2   = x3 + GRAPHS * 336;              // 264*128
    float* wp3   = wp2 + 264 * 128;                // 336*128

    const int prepN = 264 * 128 + 336 * 128 + GRAPHS + 3 * GRAPHS;
    prep_kernel<<<(prepN + 255) / 256, 256, 0, stream>>>(ec2_w1, ec3_w1, wp2, wp3, x2, x3);

    seg_mean_kernel<<<GRAPHS, 256, 0, stream>>>(node, batch, n_nodes, gmean);

    // g = GELU(gmean @ pool_w + pool_b): M=1024,N=256,K=256 -> 1024 tiles / 8 waves
    wmma_gemm_f32<<<128, 256, 0, stream>>>(gmean, HID, pool_w, pool_b,
                                           out_g, HID, x2, 264, x3, 336,
                                           GRAPHS, HID, HID, 1);
    // h1 = GELU(g @ ec1_w1 + b): M=1024,N=128,K=256 -> 512 tiles
    wmma_gemm_f32<<<64, 256, 0, stream>>>(out_g, HID, ec1_w1, ec1_b1,
                                          hbuf, 128, nullptr, 0, nullptr, 0,
                                          GRAPHS, 128, HID, 1);
    head_kernel<<<GRAPHS, 128, 0, stream>>>(hbuf, ec1_w2, ec1_b2, 7,
                                            out_ec1, 7, 1, out_p1, 7,
                                            x2, 264, 256, x3, 336, 256);
    // h2 = GELU(x2 @ wp2 + b): K=264 (padded from 263)
    wmma_gemm_f32<<<64, 256, 0, stream>>>(x2, 264, wp2, ec2_b1,
                                          hbuf, 128, nullptr, 0, nullptr, 0,
                                          GRAPHS, 128, 264, 1);
    head_kernel<<<GRAPHS, 128, 0, stream>>>(hbuf, ec2_w2, ec2_b2, 70,
                                            out_ec2, 70, 1, nullptr, 0,
                                            x3, 336, 263, nullptr, 0, 0);
    // h3 = GELU(x3 @ wp3 + b): K=336 (padded from 333)
    wmma_gemm_f32<<<64, 256, 0, stream>>>(x3, 336, wp3, ec3_b1,
                                          hbuf, 128, nullptr, 0, nullptr, 0,
                                          GRAPHS, 128, 336, 1);
    head_kernel<<<GRAPHS, 128, 0, stream>>>(hbuf, ec3_w2, ec3_b2, 300,
                                            out_ec3, 300, 0, nullptr, 0,
                                            nullptr, 0, 0, nullptr, 0, 0);
}